// Render_RGB_32091995635814
// MI455X (gfx1250) — compile-verified
//
#include <hip/hip_runtime.h>
#include <stdint.h>

#define B_   16
#define H_   512
#define W_   512
#define NV_  35709
#define NT_  70000
#define HW_  (H_ * W_)

// ---------------------------------------------------------------------------
// Pass B: per-vertex shaded color = texture * nbl * vis
// ---------------------------------------------------------------------------
__global__ __launch_bounds__(256) void shade_kernel(
    const float* __restrict__ tex, const float* __restrict__ nbl,
    const int*   __restrict__ vis, float* __restrict__ shaded)
{
    int i = blockIdx.x * 256 + threadIdx.x;          // flat over B*NV
    if (i >= B_ * NV_) return;
    float v = (float)vis[i];
    int j = i * 3;
    shaded[j + 0] = tex[j + 0] * nbl[j + 0] * v;
    shaded[j + 1] = tex[j + 1] * nbl[j + 1] * v;
    shaded[j + 2] = tex[j + 2] * nbl[j + 2] * v;
}

// ---------------------------------------------------------------------------
// Pass C: one thread per (batch, triangle). Stage this block's 256 triangles
// of tri_inds (768 dwords = 3 KB) into LDS with CDNA5 async global->LDS
// copies (ASYNCcnt path), then gather vertices and scatter-add with atomics.
// ---------------------------------------------------------------------------
__global__ __launch_bounds__(256) void tri_splat_kernel(
    const float* __restrict__ proj_geo,   // (B,NV,3)
    const float* __restrict__ shaded,     // (B,NV,3)
    const int*   __restrict__ is_visible, // (B,NV)
    const int*   __restrict__ tri_inds,   // (NT,3)
    float* __restrict__ csum,             // (B,HW,3)
    float* __restrict__ wsum)             // (B,HW)
{
    __shared__ int sh[768];

    const int b    = blockIdx.y;
    const int t0   = blockIdx.x * 256;     // first triangle handled by block
    const int thr  = threadIdx.x;
    const int base3 = t0 * 3;              // first dword of the chunk

    // --- async stage tri_inds chunk into LDS (gfx1250 ASYNCcnt path) -------
#pragma unroll
    for (int k = 0; k < 3; ++k) {
        int d = thr + k * 256;             // dword within chunk [0,768)
        int g = base3 + d;                 // global dword index
        if (g < NT_ * 3) {
            unsigned   lds_addr = (unsigned)(size_t)(&sh[d]); // low 32b of flat = LDS byte addr
            const int* gptr     = tri_inds + g;
            asm volatile("global_load_async_to_lds_b32 %0, %1, off"
                         :: "v"(lds_addr), "v"(gptr) : "memory");
        }
    }
    asm volatile("s_wait_asynccnt 0x0" ::: "memory");
    __syncthreads();

    const int t = t0 + thr;
    if (t >= NT_) return;

    const int i0 = sh[thr * 3 + 0];
    const int i1 = sh[thr * 3 + 1];
    const int i2 = sh[thr * 3 + 2];

    const int vb = b * NV_;
    const float v0 = (float)is_visible[vb + i0];
    const float v1 = (float)is_visible[vb + i1];
    const float v2 = (float)is_visible[vb + i2];
    const float w  = fminf(fminf(v0, v1), v2);       // tri_w = min vertex vis
    if (!(w > 0.0f)) return;                         // contribution is exactly 0

    const float* pg = proj_geo + (size_t)b * NV_ * 3;
    const float* sc = shaded   + (size_t)b * NV_ * 3;

    // centroid (only x,y are consumed downstream)
    const float mx = (pg[i0 * 3 + 0] + pg[i1 * 3 + 0] + pg[i2 * 3 + 0]) / 3.0f;
    const float my = (pg[i0 * 3 + 1] + pg[i1 * 3 + 1] + pg[i2 * 3 + 1]) / 3.0f;

    // mean shaded color
    const float c0 = (sc[i0 * 3 + 0] + sc[i1 * 3 + 0] + sc[i2 * 3 + 0]) / 3.0f;
    const float c1 = (sc[i0 * 3 + 1] + sc[i1 * 3 + 1] + sc[i2 * 3 + 1]) / 3.0f;
    const float c2 = (sc[i0 * 3 + 2] + sc[i1 * 3 + 2] + sc[i2 * 3 + 2]) / 3.0f;

    // round-half-even (matches jnp.round), then clip
    const int ix = (int)fminf(fmaxf(rintf(mx), 0.0f), (float)(W_ - 1));
    const int iy = (int)fminf(fmaxf(rintf(my), 0.0f), (float)(H_ - 1));

    const size_t lin = (size_t)b * HW_ + (size_t)iy * W_ + ix;
    atomicAdd(&csum[lin * 3 + 0], c0 * w);
    atomicAdd(&csum[lin * 3 + 1], c1 * w);
    atomicAdd(&csum[lin * 3 + 2], c2 * w);
    atomicAdd(&wsum[lin], w);
}

// ---------------------------------------------------------------------------
// Pass D: compose. One thread handles 4 consecutive pixels (12 floats) so all
// traffic is 16B-aligned b128 loads/stores. Streams ~233 MB -> bandwidth bound.
// ---------------------------------------------------------------------------
__global__ __launch_bounds__(256) void compose_kernel(
    const float* __restrict__ csum,   // (B*HW,3)
    const float* __restrict__ wsum,   // (B*HW)
    const int*   __restrict__ pvalid, // (B*HW)
    const float* __restrict__ ori,    // (B*HW,3)
    float* __restrict__ render,       // (B*HW,3)
    float* __restrict__ realout)      // (B*HW,3)
{
    const size_t g = (size_t)blockIdx.x * 256 + threadIdx.x;  // pixel group
    const size_t ngroups = (size_t)B_ * HW_ / 4;
    if (g >= ngroups) return;
    const size_t p = g * 4;                                   // first pixel

    const float4 w4  = *(const float4*)(wsum + p);
    const int4   pv4 = *(const int4*)(pvalid + p);

    float4 cv[3], ov[3];
    const float4* cp = (const float4*)(csum + p * 3);
    const float4* op = (const float4*)(ori  + p * 3);
    cv[0] = cp[0]; cv[1] = cp[1]; cv[2] = cp[2];
    ov[0] = op[0]; ov[1] = op[1]; ov[2] = op[2];

    const float* c = (const float*)cv;
    const float* o = (const float*)ov;
    const float  wv[4] = {w4.x, w4.y, w4.z, w4.w};
    const int    pvv[4] = {pv4.x, pv4.y, pv4.z, pv4.w};

    float4 rn[3], rl[3];
    float* rn_s = (float*)rn;
    float* rl_s = (float*)rl;

#pragma unroll
    for (int j = 0; j < 4; ++j) {
        const float w      = wv[j];
        const float pvf    = (float)pvv[j];
        const bool  valid  = pvv[j] > 0;
        const bool  cover  = (w > 0.0f) && valid;
        const float inv    = 1.0f / fmaxf(w, 1.0f);
#pragma unroll
        for (int ch = 0; ch < 3; ++ch) {
            const float oc   = o[j * 3 + ch];
            const float face = c[j * 3 + ch] * inv;
            rn_s[j * 3 + ch] = cover ? face : oc;
            rl_s[j * 3 + ch] = oc * pvf;           // oi * pixel_valid
        }
    }

    float4* rnp = (float4*)(render  + p * 3);
    float4* rlp = (float4*)(realout + p * 3);
    rnp[0] = rn[0]; rnp[1] = rn[1]; rnp[2] = rn[2];
    rlp[0] = rl[0]; rlp[1] = rl[1]; rlp[2] = rl[2];
}

// ---------------------------------------------------------------------------
extern "C" void kernel_launch(void* const* d_in, const int* in_sizes, int n_in,
                              void* d_out, int out_size, void* d_ws, size_t ws_size,
                              hipStream_t stream)
{
    const float* proj_geo  = (const float*)d_in[0];  // (B,NV,3)
    const float* texture   = (const float*)d_in[1];  // (B,NV,3)
    const float* nbl       = (const float*)d_in[2];  // (B,NV,3)
    const float* ori_img   = (const float*)d_in[3];  // (B,H,W,3)
    const int*   is_vis    = (const int*)d_in[4];    // (B,NV)
    const int*   tri_inds  = (const int*)d_in[5];    // (NT,3)
    const int*   pix_valid = (const int*)d_in[6];    // (B,H,W)

    // workspace layout: [csum | wsum] (zeroed each call) then [shaded]
    const size_t CS = (size_t)B_ * HW_ * 3;   // 12,582,912 floats
    const size_t WS = (size_t)B_ * HW_;       //  4,194,304 floats
    float* csum   = (float*)d_ws;
    float* wsum   = csum + CS;
    float* shaded = wsum + WS;

    hipMemsetAsync(d_ws, 0, (CS + WS) * sizeof(float), stream);

    // Pass B: shade vertices
    {
        const int n = B_ * NV_;
        shade_kernel<<<(n + 255) / 256, 256, 0, stream>>>(texture, nbl, is_vis, shaded);
    }

    // Pass C: triangle scatter (grid.y = batch)
    {
        dim3 grid((NT_ + 255) / 256, B_);
        tri_splat_kernel<<<grid, 256, 0, stream>>>(proj_geo, shaded, is_vis,
                                                   tri_inds, csum, wsum);
    }

    // Pass D: compose render + real
    {
        float* render  = (float*)d_out;
        float* realout = render + CS;
        const size_t ngroups = (size_t)B_ * HW_ / 4;   // 1,048,576
        compose_kernel<<<(unsigned)((ngroups + 255) / 256), 256, 0, stream>>>(
            csum, wsum, pix_valid, ori_img, render, realout);
    }
}